// NonZeroAvgPool_79843442032848
// MI455X (gfx1250) — compile-verified
//
#include <hip/hip_runtime.h>

// NonZeroAvgPool for MI455X (gfx1250, wave32):
//   out[b,d] = sum_s (tokens[b,s]!=0) * x[b,s,d] / count_valid[b]
// Bandwidth-bound (134 MB @ 23.3 TB/s ~= 5.8 us). Reduction is done on the
// matrix pipe with V_WMMA_F32_16X16X4_F32: A = broadcast valid mask (all 16
// rows identical), B = 4x16 tile of x, C accumulates masked column sums.

typedef float v2f __attribute__((ext_vector_type(2)));
typedef float v4f __attribute__((ext_vector_type(4)));
typedef float v8f __attribute__((ext_vector_type(8)));

#define PAD_ID 0
#define S_LEN 4096
#define D_LEN 512
#define WAVES 16                       // 512 threads / 32
#define ROWS_PER_WAVE (S_LEN / WAVES)  // 256
#define CHUNKS (ROWS_PER_WAVE / 4)     // 64 WMMA K-chunks per wave

__global__ __launch_bounds__(512, 1)
void nz_avg_pool_wmma(const float* __restrict__ x,
                      const int* __restrict__ tokens,
                      float* __restrict__ out) {
  const int dt   = blockIdx.x;   // 0..7  : 64-column d-tile
  const int b    = blockIdx.y;   // 0..15 : batch
  const int d0   = dt * 64;
  const int tid  = threadIdx.x;
  const int w    = tid >> 5;     // wave in block
  const int lane = tid & 31;
  const int n    = lane & 15;            // column within 16-wide WMMA tile
  const int kA   = (lane & 16) ? 2 : 0;  // K half owned by this lane group

  const float* __restrict__ xb = x + (size_t)b * S_LEN * D_LEN;
  const int*   __restrict__ tb = tokens + (size_t)b * S_LEN;

  const int sw = w * ROWS_PER_WAVE;

  v8f acc0 = {}, acc1 = {}, acc2 = {}, acc3 = {};
  float cnt = 0.0f;

  // lane streams rows (s+kA, s+kA+1), 4 consecutive cols at d0 + 4n
  const float* px = xb + (size_t)(sw + kA) * D_LEN + d0 + 4 * n;
  const int*   pt = tb + sw + kA;

#pragma unroll 2
  for (int i = 0; i < CHUNKS; ++i) {
    // x tile: 4 rows x 64 cols for this wave's chunk (b128, fully coalesced,
    // non-temporal: streamed exactly once)
    v4f r0 = __builtin_nontemporal_load((const v4f*)px);          // row s+kA
    v4f r1 = __builtin_nontemporal_load((const v4f*)(px + D_LEN));// row s+kA+1
    px += 4 * D_LEN;

    // token pair for this lane's two rows (8B aligned: s%4==0, kA even)
    int2 t = *(const int2*)pt;
    pt += 4;
    float a0 = (t.x != PAD_ID) ? 1.0f : 0.0f;
    float a1 = (t.y != PAD_ID) ? 1.0f : 0.0f;
    // count each s-row once: only column-0 lanes (0 and 16) contribute
    cnt += (n == 0) ? (a0 + a1) : 0.0f;

    // A (16x4): valid mask broadcast to all 16 rows.
    // Lane layout: VGPR0 = K=kA, VGPR1 = K=kA+1  (same K convention as B)
    v2f a = {a0, a1};

    // B (4x16): VGPR0 = row kA, VGPR1 = row kA+1; the 4 floats of the b128
    // feed 4 independent tiles (memory col = d0 + 4n + e for tile e)
    v2f b0 = {r0.x, r1.x};
    v2f b1 = {r0.y, r1.y};
    v2f b2 = {r0.z, r1.z};
    v2f b3 = {r0.w, r1.w};

    acc0 = __builtin_amdgcn_wmma_f32_16x16x4_f32(false, a, false, b0,
                                                 (short)0, acc0, false, false);
    acc1 = __builtin_amdgcn_wmma_f32_16x16x4_f32(false, a, false, b1,
                                                 (short)0, acc1, false, false);
    acc2 = __builtin_amdgcn_wmma_f32_16x16x4_f32(false, a, false, b2,
                                                 (short)0, acc2, false, false);
    acc3 = __builtin_amdgcn_wmma_f32_16x16x4_f32(false, a, false, b3,
                                                 (short)0, acc3, false, false);
  }

  // Every row of each C tile holds the same partial column sums; take row M=0
  // (VGPR0 on lanes 0..15) and combine the 16 waves through LDS.
  __shared__ float sAcc[WAVES][4][16];  // 4 KB
  __shared__ float sCnt[WAVES];

  if (lane < 16) {
    sAcc[w][0][n] = acc0[0];
    sAcc[w][1][n] = acc1[0];
    sAcc[w][2][n] = acc2[0];
    sAcc[w][3][n] = acc3[0];
  }
  // wave count: lane0 holds K={0,1} rows, lane16 holds K={2,3} rows
  float cw = cnt + __shfl_xor(cnt, 16, 32);
  if (lane == 0) sCnt[w] = cw;
  __syncthreads();

  // 64 output columns of this tile; col offset = 4*n + e
  if (tid < 64) {
    const int e  = tid & 3;
    const int nn = tid >> 2;
    float tot = 0.0f, c = 0.0f;
    for (int ww = 0; ww < WAVES; ++ww) {
      tot += sAcc[ww][e][nn];
      c   += sCnt[ww];
    }
    out[(size_t)b * D_LEN + d0 + tid] = tot / c;
  }
}

extern "C" void kernel_launch(void* const* d_in, const int* in_sizes, int n_in,
                              void* d_out, int out_size, void* d_ws,
                              size_t ws_size, hipStream_t stream) {
  (void)in_sizes; (void)n_in; (void)d_ws; (void)ws_size; (void)out_size;
  const float* x      = (const float*)d_in[0];
  const int*   tokens = (const int*)d_in[1];
  float*       out    = (float*)d_out;

  dim3 grid(D_LEN / 64, 16);  // (8 d-tiles, 16 batches)
  nz_avg_pool_wmma<<<grid, 512, 0, stream>>>(x, tokens, out);
}